// RegionLoss_43267500540733
// MI455X (gfx1250) — compile-verified
//
#include <hip/hip_runtime.h>

// ---------------------------------------------------------------------------
// Region loss for MI455X (gfx1250, wave32).
// Pairwise keypoint distance stage mapped onto V_WMMA_F32_16X16X4_F32:
//   d^2(t,m,k) = u_t(k) . v_m(k),  u=[-2ax,-2ay,ax^2+ay^2,1], v=[bx,by,1,bx^2+by^2]
// Fast-path transcendentals: raw v_sqrt_f32 / v_rcp_f32 / v_exp_f32 (exp2).
// cornerf branch removed: f(d) = max(exp2(fma(d,-log2e/15,2log2e))*INV-INV, 0)
// which equals the reference's gated expression exactly (<=0 for d>=30).
// Element loop is staged 8-wide so the TRANS32 chains interleave (no v_nops).
// ---------------------------------------------------------------------------

typedef __attribute__((ext_vector_type(2))) float v2f;
typedef __attribute__((ext_vector_type(8))) float v8f;

#define KP      9
#define NB      32
#define NANCH   5
#define NHW     32
#define NPIX    1024            // 32*32
#define MCELL   5120            // NANCH*NPIX
#define NCELL   163840          // NB*MCELL
#define NCLS    13
#define CSTRIDE 1024            // channel stride in floats (nH*nW)

#define INVDEN   0.15651742f    // 1/(e^2 - 1 + 1e-5)
#define NL2E15   -0.0961796694f // -log2(e)/15
#define TWOL2E   2.8853900818f  // 2*log2(e)

__device__ __forceinline__ float fsqrt_raw(float x) { return __builtin_amdgcn_sqrtf(x); }
__device__ __forceinline__ float frcp_raw(float x)  { return __builtin_amdgcn_rcpf(x); }
__device__ __forceinline__ float fexp2_raw(float x) { return __builtin_amdgcn_exp2f(x); }

__device__ __forceinline__ float sigf(float x) {
  return frcp_raw(1.f + fexp2_raw(x * -1.4426950409f));   // v_mul+v_exp+v_rcp
}

// Reference: (exp(2*(1-d/30)) - 1) / (e^2-1+1e-5) for d < 30 else 0.
__device__ __forceinline__ float cornerf(float d) {
  float e = fexp2_raw(__builtin_fmaf(d, NL2E15, TWOL2E));
  return fmaxf(__builtin_fmaf(e, INVDEN, -INVDEN), 0.f);
}

__device__ __forceinline__ int chx_of(int k) { return k ? (k + 2) : 0; }
__device__ __forceinline__ int chy_of(int k) { return k ? (k + 3) : 1; }

// ---------------------------------------------------------------------------
__global__ __launch_bounds__(256)
void init_kernel(int* __restrict__ tmap, float* __restrict__ out_scalar) {
  int n = blockIdx.x * 256 + threadIdx.x;
  if (n < NCELL) tmap[n] = 0;
  if (n == 0) out_scalar[0] = 0.f;
}

// ---------------------------------------------------------------------------
// One thread per (b, t), t in [0,64): target-side precompute.
__global__ __launch_bounds__(64)
void prep_targets(const float* __restrict__ tgt, const float* __restrict__ out,
                  float4* __restrict__ TA, float* __restrict__ VALID,
                  float* __restrict__ TXS, float* __restrict__ TYS,
                  float* __restrict__ CONF_T, int* __restrict__ TCLS,
                  int* __restrict__ TMAP) {
  int b = blockIdx.x;
  int t = threadIdx.x;

  if (t >= 50) {                      // pad rows for the 4x16 WMMA target tiles
    VALID[b * 64 + t] = 0.f;
#pragma unroll
    for (int k = 0; k < KP; ++k)
      TA[(b * 64 + t) * KP + k] = make_float4(0.f, 0.f, 0.f, 1.f);
    return;
  }

  const float* row = tgt + (size_t)b * 1050 + t * 21;
  float gxn[KP], gyn[KP];
#pragma unroll
  for (int k = 0; k < KP; ++k) { gxn[k] = row[1 + 2 * k]; gyn[k] = row[2 + 2 * k]; }

  // valid = cumprod over t' <= t of (gxn[t'][0] != 0)
  bool valid = true;
  for (int u = 0; u <= t; ++u)
    valid = valid && (tgt[(size_t)b * 1050 + u * 21 + 1] != 0.f);
  VALID[b * 64 + t] = valid ? 1.f : 0.f;

  // A-operand 4-vectors (pixel space)
#pragma unroll
  for (int k = 0; k < KP; ++k) {
    float ax = 640.f * gxn[k], ay = 480.f * gyn[k];
    TA[(b * 64 + t) * KP + k] = make_float4(-2.f * ax, -2.f * ay, ax * ax + ay * ay, 1.f);
  }

  float gx0 = gxn[0] * 32.f, gy0 = gyn[0] * 32.f;
  int gi0 = (int)gx0, gj0 = (int)gy0;
#pragma unroll
  for (int k = 0; k < KP; ++k) {
    TXS[(b * 50 + t) * KP + k] = gxn[k] * 32.f - (float)gi0;
    TYS[(b * 50 + t) * KP + k] = gyn[k] * 32.f - (float)gj0;
  }
  int cls = (int)row[0];
  TCLS[b * 50 + t] = (cls < 0) ? 0 : ((cls > NCLS - 1) ? NCLS - 1 : cls);

  // anchor IoU -> best_n (first max wins; -1 -> python (-1 % 5) == 4)
  float gw = row[19] * 32.f, gh = row[20] * 32.f;
  const float aw[5] = {1.482f, 2.0501f, 2.3946f, 3.1018f, 3.4879f};
  const float ah[5] = {2.2412f, 3.1265f, 4.6891f, 3.991f, 5.2505f};
  int best = -1; float bi = 0.f;
#pragma unroll
  for (int a = 0; a < 5; ++a) {
    float cw = fminf(aw[a], gw), chh = fminf(ah[a], gh);
    float inter = cw * chh;
    float uni = aw[a] * ah[a] + gw * gh - inter;
    float iou = (cw > 0.f && chh > 0.f) ? inter / uni : 0.f;
    if (iou > bi) { bi = iou; best = a; }
  }
  int an = (best < 0) ? 4 : best;

  int flat = ((b * NANCH + an) * NHW + gj0) * NHW + gi0;
  if (valid) atomicMax(&TMAP[flat], t + 1);

  // pb with python floored mod (negative for b==0)
  long long ll = (long long)b * MCELL - NPIX + (long long)gj0 * NHW + gi0;
  long long Nn = NCELL;
  int pb = (int)(((ll % Nn) + Nn) % Nn);
  int b2 = pb / MCELL, m2 = pb % MCELL;
  int a2 = m2 >> 10, j2 = (m2 >> 5) & 31, i2 = m2 & 31;
  const float* ob = out + ((((size_t)b2 * 160 + a2 * 32) * 32 + j2) * 32 + i2);

  float csum = 0.f;
#pragma unroll
  for (int k = 0; k < KP; ++k) {
    float xv = ob[(size_t)chx_of(k) * CSTRIDE];
    float yv = ob[(size_t)chy_of(k) * CSTRIDE];
    if (k == 0) { xv = sigf(xv); yv = sigf(yv); }
    float px = (xv + (float)i2) * (1.f / 32.f);
    float py = (yv + (float)j2) * (1.f / 32.f);
    float dx = (gxn[k] - px) * 640.f;
    float dy = (gyn[k] - py) * 480.f;
    csum += cornerf(fsqrt_raw(dx * dx + dy * dy));
  }
  CONF_T[b * 50 + t] = csum * (1.f / (float)KP);
}

// ---------------------------------------------------------------------------
// WMMA kernel: 8 waves/block, each wave owns one 16-cell tile of one batch.
// cur[n] = max over valid targets of mean_k f(d(t,n,k))
__global__ __launch_bounds__(256)
void conf_max_kernel(const float* __restrict__ out, const float4* __restrict__ TA,
                     const float* __restrict__ VALID, float* __restrict__ CUR) {
  int lane = threadIdx.x & 31;
  int warp = threadIdx.x >> 5;
  int tile = blockIdx.x * 8 + warp;      // 0..319
  int b    = blockIdx.y;                 // 0..31
  int ci = lane & 15;                    // cell within tile / A-row within tile
  int hi = lane >> 4;                    // lane half

  int m = tile * 16 + ci;                // cell within batch
  int a = m >> 10, j = (m >> 5) & 31, i = m & 31;
  const float* ob = out + ((((size_t)b * 160 + a * 32) * 32 + j) * 32 + i);

  // Build 9 B fragments (4x16 f32): VGPR0 = B[0]/B[2], VGPR1 = B[1]/B[3]
  v2f Bf[KP];
#pragma unroll
  for (int k = 0; k < KP; ++k) {
    float xv = ob[(size_t)chx_of(k) * CSTRIDE];
    float yv = ob[(size_t)chy_of(k) * CSTRIDE];
    if (k == 0) { xv = sigf(xv); yv = sigf(yv); }
    float bx = 20.f * (xv + (float)i);   // 640 * (xs+i)/32
    float by = 15.f * (yv + (float)j);   // 480 * (ys+j)/32
    v2f f;
    f.x = hi ? 1.f : bx;
    f.y = hi ? (bx * bx + by * by) : by;
    Bf[k] = f;
  }

  const v8f czero = {0.f, 0.f, 0.f, 0.f, 0.f, 0.f, 0.f, 0.f};
  float runmax = 0.f;

#pragma unroll
  for (int tt = 0; tt < 4; ++tt) {       // 4 tiles of 16 targets (50 padded to 64)
    float csum[8] = {0.f, 0.f, 0.f, 0.f, 0.f, 0.f, 0.f, 0.f};
#pragma unroll
    for (int k = 0; k < KP; ++k) {
      int ta = tt * 16 + ci;             // A row this lane supplies
      float4 u = TA[((size_t)b * 64 + ta) * KP + k];
      v2f Af;
      Af.x = hi ? u.z : u.x;
      Af.y = hi ? u.w : u.y;
      // D = A(16x4) x B(4x16) : 16x16 tile of squared pixel distances
      v8f d = __builtin_amdgcn_wmma_f32_16x16x4_f32(
          false, Af, false, Bf[k], (short)0, czero, false, false);
      // Staged 8-wide so the sqrt/exp chains interleave (hide TRANS latency).
      float e[8];
#pragma unroll
      for (int r = 0; r < 8; ++r)        // |.| folds into v_sqrt source modifier
        e[r] = fsqrt_raw(__builtin_fabsf(d[r]));
#pragma unroll
      for (int r = 0; r < 8; ++r)
        e[r] = fexp2_raw(__builtin_fmaf(e[r], NL2E15, TWOL2E));
#pragma unroll
      for (int r = 0; r < 8; ++r)
        csum[r] += fmaxf(__builtin_fmaf(e[r], INVDEN, -INVDEN), 0.f);
    }
    // valid mask + max over this tile's 16 rows (8 per lane half)
#pragma unroll
    for (int r = 0; r < 8; ++r) {
      int t = tt * 16 + r + 8 * hi;      // C/D layout: M = r + 8*(lane>=16)
      float v = VALID[b * 64 + t];
      float c = csum[r] * (1.f / (float)KP) * v;
      runmax = fmaxf(runmax, c);
    }
  }
  // merge the two lane halves (same column N = lane&15, disjoint rows)
  runmax = fmaxf(runmax, __shfl_xor(runmax, 16, 32));
  if (hi == 0) CUR[(size_t)b * MCELL + m] = runmax;
}

// ---------------------------------------------------------------------------
__global__ __launch_bounds__(256)
void final_loss(const float* __restrict__ out, const int* __restrict__ TMAP,
                const float* __restrict__ CUR, const float* __restrict__ TXS,
                const float* __restrict__ TYS, const float* __restrict__ CONF_T,
                const int* __restrict__ TCLS, const int* __restrict__ epochp,
                float* __restrict__ out_scalar) {
  int n = blockIdx.x * 256 + threadIdx.x;
  float lsum = 0.f;
  if (n < NCELL) {
    int b = n / MCELL, m = n % MCELL;
    int a = m >> 10, j = (m >> 5) & 31, i = m & 31;
    const float* ob = out + ((((size_t)b * 160 + a * 32) * 32 + j) * 32 + i);
    int epoch = *epochp;
    int hc = TMAP[n];
    float cf = sigf(ob[(size_t)18 * CSTRIDE]);

    if (hc > 0) {
      int tb = b * 50 + (hc - 1);
      float acc = 0.f;
#pragma unroll
      for (int k = 0; k < KP; ++k) {
        float xv = ob[(size_t)chx_of(k) * CSTRIDE];
        float yv = ob[(size_t)chy_of(k) * CSTRIDE];
        if (k == 0) { xv = sigf(xv); yv = sigf(yv); }
        float dx = xv - TXS[(size_t)tb * KP + k];
        float dy = yv - TYS[(size_t)tb * KP + k];
        acc += dx * dx + dy * dy;
      }
      lsum = 0.5f * acc;                                   // COORD*0.5*(lx+ly)

      // class loss: logsumexp - picked
      float lg[NCLS];
      float mx = -3.402823466e38f;
#pragma unroll
      for (int c = 0; c < NCLS; ++c) {
        lg[c] = ob[(size_t)(19 + c) * CSTRIDE];
        mx = fmaxf(mx, lg[c]);
      }
      float s = 0.f;
#pragma unroll
      for (int c = 0; c < NCLS; ++c) s += __expf(lg[c] - mx);
      float lse = mx + __logf(s);
      lsum += lse - lg[TCLS[tb]];

      if (epoch > 15) {                                    // obj conf loss
        float dcf = cf - CONF_T[tb];
        lsum += 0.5f * 5.0f * dcf * dcf;                   // OBJ = 5
      }
    } else {
      if (epoch > 15) {                                    // noobj conf loss
        float cm0 = (CUR[n] > 0.6f) ? 0.f : 1.f;           // SIL / NOOBJ
        lsum += 0.5f * cm0 * cf * cf;
      }
    }
  }
  // wave32 reduce, one atomic per wave
#pragma unroll
  for (int off = 16; off > 0; off >>= 1) lsum += __shfl_down(lsum, off, 32);
  if ((threadIdx.x & 31) == 0) atomicAdd(out_scalar, lsum);
}

// ---------------------------------------------------------------------------
extern "C" void kernel_launch(void* const* d_in, const int* in_sizes, int n_in,
                              void* d_out, int out_size, void* d_ws, size_t ws_size,
                              hipStream_t stream) {
  const float* out_t = (const float*)d_in[0];   // (32,160,32,32) f32
  const float* tgt   = (const float*)d_in[1];   // (32,1050) f32
  const int*   epoch = (const int*)d_in[2];     // scalar
  float* loss = (float*)d_out;

  char* ws = (char*)d_ws;
  size_t off = 0;
  auto alloc = [&](size_t bytes) -> void* {
    off = (off + 255) & ~(size_t)255;
    void* p = ws + off;
    off += bytes;
    return p;
  };

  int*    TMAP   = (int*)   alloc((size_t)NCELL * sizeof(int));
  float*  CUR    = (float*) alloc((size_t)NCELL * sizeof(float));
  float4* TA     = (float4*)alloc((size_t)NB * 64 * KP * sizeof(float4));
  float*  VALID  = (float*) alloc((size_t)NB * 64 * sizeof(float));
  float*  TXS    = (float*) alloc((size_t)NB * 50 * KP * sizeof(float));
  float*  TYS    = (float*) alloc((size_t)NB * 50 * KP * sizeof(float));
  float*  CONF_T = (float*) alloc((size_t)NB * 50 * sizeof(float));
  int*    TCLS   = (int*)   alloc((size_t)NB * 50 * sizeof(int));
  (void)ws_size; (void)in_sizes; (void)n_in; (void)out_size;

  init_kernel<<<(NCELL + 255) / 256, 256, 0, stream>>>(TMAP, loss);
  prep_targets<<<NB, 64, 0, stream>>>(tgt, out_t, TA, VALID, TXS, TYS, CONF_T,
                                      TCLS, TMAP);
  conf_max_kernel<<<dim3(40, NB), 256, 0, stream>>>(out_t, TA, VALID, CUR);
  final_loss<<<(NCELL + 255) / 256, 256, 0, stream>>>(out_t, TMAP, CUR, TXS, TYS,
                                                      CONF_T, TCLS, epoch, loss);
}